// VehicleEmbedding_34368328302902
// MI455X (gfx1250) — compile-verified
//
#include <hip/hip_runtime.h>
#include <hip/hip_bf16.h>

#define D_MODEL 512
#define GD      2048          // 4 * D_MODEL
#define N_NODES 1000
#define T_DIM   1023
#define IN_DIM  2024          // N_NODES + 1 + T_DIM
#define KK      16
#define CAP_INV (1.0f / 100.0f)
#define LN_EPS  1e-5f

// CDNA5 async global->LDS path (gfx1250). Existence of the builtin was
// proven by the round-1 type diagnostic; guard kept for portability.
#if defined(__HIP_DEVICE_COMPILE__) && \
    __has_builtin(__builtin_amdgcn_global_load_async_to_lds_b128) && \
    __has_builtin(__builtin_amdgcn_s_wait_asynccnt)
#define USE_ASYNC_LDS 1
#else
#define USE_ASYNC_LDS 0
#endif

#define AS1 __attribute__((address_space(1)))
#define AS3 __attribute__((address_space(3)))

typedef int v4i_ __attribute__((vector_size(16)));

__device__ __forceinline__ float sigmoidf_(float x) {
  return 1.0f / (1.0f + expf(-x));
}

// ---------------------------------------------------------------------------
// Main kernel: one block (8 wave32) per batch element b. Stage the shared
// time/load feature columns via CDNA5 async global->LDS DMA, then for each of
// the K=16 vehicles gather the pos column (coalesced, L2-resident), compute
// the 3 live LSTM gates (f-gate is dead since c0=0), LayerNorm, stream out.
// ---------------------------------------------------------------------------
__global__ __launch_bounds__(256) void vehemb_kernel(
    const float* __restrict__ WT, const float* __restrict__ W_ih,
    const int* __restrict__ positions, const float* __restrict__ loads,
    const int* __restrict__ timev, const float* __restrict__ b_ih,
    const float* __restrict__ b_hh, const float* __restrict__ gamma,
    const float* __restrict__ beta, float* __restrict__ out, const int useWT) {
  __shared__ __align__(16) float ldsT[GD];   // time column  W_T[1001+t][:]
  __shared__ __align__(16) float ldsL[GD];   // load column  W_T[1000][:]
  __shared__ __align__(16) float ldsB[GD];   // b_ih + b_hh
  __shared__ float redS[8];
  __shared__ float redQ[8];

  const int tid = threadIdx.x;
  const int b = blockIdx.x;
  int t = timev[b];
  t = t < 0 ? 0 : (t > T_DIM - 1 ? T_DIM - 1 : t);
  const int ft = N_NODES + 1 + t;   // feature index of time one-hot

  if (useWT) {
    const float* tc = WT + (size_t)ft * GD;
    const float* lc = WT + (size_t)N_NODES * GD;
#if USE_ASYNC_LDS
    // Async DMA the two shared 8KB columns into LDS (tracked by ASYNCcnt).
    for (int i = tid; i < GD / 4; i += 256) {
      __builtin_amdgcn_global_load_async_to_lds_b128(
          (AS1 v4i_*)(tc + i * 4), (AS3 v4i_*)(ldsT + i * 4), 0, 0);
      __builtin_amdgcn_global_load_async_to_lds_b128(
          (AS1 v4i_*)(lc + i * 4), (AS3 v4i_*)(ldsL + i * 4), 0, 0);
    }
#else
    for (int i = tid; i < GD; i += 256) {
      ldsT[i] = tc[i];
      ldsL[i] = lc[i];
    }
#endif
  } else {
    // Fallback: workspace too small for W_T; read strided columns of W_ih.
    for (int i = tid; i < GD; i += 256) {
      ldsT[i] = W_ih[(size_t)i * IN_DIM + ft];
      ldsL[i] = W_ih[(size_t)i * IN_DIM + N_NODES];
    }
  }
  for (int i = tid; i < GD; i += 256) ldsB[i] = b_ih[i] + b_hh[i];
#if USE_ASYNC_LDS
  if (useWT) __builtin_amdgcn_s_wait_asynccnt(0);
#endif
  __syncthreads();

  // Each thread owns output dims d0, d1. All staged terms are k-invariant:
  // pull them into registers once.
  const int d0 = tid, d1 = tid + 256;
  const float tb_i0 = ldsT[d0] + ldsB[d0];
  const float tb_i1 = ldsT[d1] + ldsB[d1];
  const float tb_g0 = ldsT[2 * D_MODEL + d0] + ldsB[2 * D_MODEL + d0];
  const float tb_g1 = ldsT[2 * D_MODEL + d1] + ldsB[2 * D_MODEL + d1];
  const float tb_o0 = ldsT[3 * D_MODEL + d0] + ldsB[3 * D_MODEL + d0];
  const float tb_o1 = ldsT[3 * D_MODEL + d1] + ldsB[3 * D_MODEL + d1];
  const float ll_i0 = ldsL[d0], ll_i1 = ldsL[d1];
  const float ll_g0 = ldsL[2 * D_MODEL + d0], ll_g1 = ldsL[2 * D_MODEL + d1];
  const float ll_o0 = ldsL[3 * D_MODEL + d0], ll_o1 = ldsL[3 * D_MODEL + d1];
  const float gm0 = gamma[d0], gm1 = gamma[d1];
  const float bt0 = beta[d0], bt1 = beta[d1];
  const int lane = tid & 31, wv = tid >> 5;

  for (int k = 0; k < KK; ++k) {
    const int p = positions[b * KK + k];
    const float ln = loads[b * KK + k] * CAP_INV;

    float pi0, pi1, pg0, pg1, po0, po1;
    if (useWT) {
      const float* pc = WT + (size_t)p * GD;
      pi0 = pc[d0];
      pi1 = pc[d1];
      pg0 = pc[2 * D_MODEL + d0];
      pg1 = pc[2 * D_MODEL + d1];
      po0 = pc[3 * D_MODEL + d0];
      po1 = pc[3 * D_MODEL + d1];
      if (k + 1 < KK) {
        // Warm caches for the next vehicle's 8KB pos column.
        const float* nx = WT + (size_t)positions[b * KK + k + 1] * GD;
        __builtin_prefetch(nx + tid * 8, 0, 1);
      }
    } else {
      pi0 = W_ih[(size_t)d0 * IN_DIM + p];
      pi1 = W_ih[(size_t)d1 * IN_DIM + p];
      pg0 = W_ih[(size_t)(2 * D_MODEL + d0) * IN_DIM + p];
      pg1 = W_ih[(size_t)(2 * D_MODEL + d1) * IN_DIM + p];
      po0 = W_ih[(size_t)(3 * D_MODEL + d0) * IN_DIM + p];
      po1 = W_ih[(size_t)(3 * D_MODEL + d1) * IN_DIM + p];
    }

    const float iv0 = fmaf(ln, ll_i0, pi0 + tb_i0);
    const float iv1 = fmaf(ln, ll_i1, pi1 + tb_i1);
    const float gv0 = fmaf(ln, ll_g0, pg0 + tb_g0);
    const float gv1 = fmaf(ln, ll_g1, pg1 + tb_g1);
    const float ov0 = fmaf(ln, ll_o0, po0 + tb_o0);
    const float ov1 = fmaf(ln, ll_o1, po1 + tb_o1);

    const float c0 = sigmoidf_(iv0) * tanhf(gv0);
    const float c1 = sigmoidf_(iv1) * tanhf(gv1);
    const float h0 = sigmoidf_(ov0) * tanhf(c0);
    const float h1 = sigmoidf_(ov1) * tanhf(c1);

    // LayerNorm over 512 dims: wave32 shfl reduce, then cross-wave via LDS.
    float s = h0 + h1;
    float q = fmaf(h0, h0, h1 * h1);
#pragma unroll
    for (int off = 16; off > 0; off >>= 1) {
      s += __shfl_xor(s, off, 32);
      q += __shfl_xor(q, off, 32);
    }
    __syncthreads();   // protect redS/redQ reuse across k iterations
    if (lane == 0) {
      redS[wv] = s;
      redQ[wv] = q;
    }
    __syncthreads();
    float S = 0.f, Q = 0.f;
#pragma unroll
    for (int w = 0; w < 8; ++w) {
      S += redS[w];
      Q += redQ[w];
    }
    const float mu = S * (1.0f / (float)D_MODEL);
    const float var = fmaf(-mu, mu, Q * (1.0f / (float)D_MODEL));
    const float rstd = 1.0f / sqrtf(var + LN_EPS);

    float* op = out + ((size_t)b * KK + k) * D_MODEL;
    op[d0] = fmaf((h0 - mu) * rstd, gm0, bt0);
    op[d1] = fmaf((h1 - mu) * rstd, gm1, bt1);
  }
}

// ---------------------------------------------------------------------------
// Tiled transpose  W_ih [GD x IN_DIM]  ->  W_T [IN_DIM x GD]
// Makes every gathered feature-column a contiguous, fully-coalesced 8KB row.
// ---------------------------------------------------------------------------
__global__ __launch_bounds__(256) void wt_transpose_kernel(
    const float* __restrict__ W, float* __restrict__ WT) {
  __shared__ float tile[32][33];
  const int f0 = blockIdx.x * 32;   // column (feature) tile
  const int g0 = blockIdx.y * 32;   // row (gate) tile
  const int tx = threadIdx.x;       // 0..31
  const int ty = threadIdx.y;       // 0..7
#pragma unroll
  for (int i = 0; i < 32; i += 8) {
    const int g = g0 + ty + i;
    const int f = f0 + tx;
    tile[ty + i][tx] = (f < IN_DIM) ? W[(size_t)g * IN_DIM + f] : 0.0f;
  }
  __syncthreads();
#pragma unroll
  for (int i = 0; i < 32; i += 8) {
    const int f = f0 + ty + i;
    const int g = g0 + tx;
    if (f < IN_DIM) WT[(size_t)f * GD + g] = tile[tx][ty + i];
  }
}

// ---------------------------------------------------------------------------
extern "C" void kernel_launch(void* const* d_in, const int* in_sizes, int n_in,
                              void* d_out, int out_size, void* d_ws,
                              size_t ws_size, hipStream_t stream) {
  const int* positions = (const int*)d_in[0];
  const float* loads   = (const float*)d_in[1];
  const int* timev     = (const int*)d_in[2];
  const float* W_ih    = (const float*)d_in[3];
  const float* b_ih    = (const float*)d_in[4];
  const float* b_hh    = (const float*)d_in[5];
  const float* gamma   = (const float*)d_in[6];
  const float* beta    = (const float*)d_in[7];
  float* out = (float*)d_out;

  const int Bn = in_sizes[2];   // batch size (time is [B])

  const size_t wtBytes = (size_t)IN_DIM * GD * sizeof(float);
  const int useWT = (d_ws != nullptr && ws_size >= wtBytes) ? 1 : 0;
  float* WT = (float*)d_ws;

  if (useWT) {
    dim3 tg((IN_DIM + 31) / 32, GD / 32);
    wt_transpose_kernel<<<tg, dim3(32, 8), 0, stream>>>(W_ih, WT);
  }
  vehemb_kernel<<<Bn, 256, 0, stream>>>(WT, W_ih, positions, loads, timev,
                                        b_ih, b_hh, gamma, beta, out, useWT);
}